// GraphBlock_7739531067618
// MI455X (gfx1250) — compile-verified
//
#include <hip/hip_runtime.h>

// ---------------------------------------------------------------------------
// GAT block, factorized.  B=2,T=64,N=64,D=64 -> P=128 graphs of 64 nodes.
//   out  = d_out[0 .. 8192)            (B,N,T) float32
//   adj  = d_out[8192 .. 8192+524288)  pre-softmax attention logits e
// WMMA:  V_WMMA_F32_16X16X4_F32 for all four GEMM families (full fp32).
// TDM:   tensor_load_to_lds stages the LDS tiles (incl. hardware stride-129
//        padding for the BV tile: pad_interval=128 DWORDs, pad_amount=1 DWORD).
// Workspace (floats): v(512K) su/squ/sv/sqv(4*8K) WpT(16K) c(128) bp(128)
//                     AU(1M) BV(1M)  -> ~10.7 MB total.
// ---------------------------------------------------------------------------

typedef __attribute__((ext_vector_type(2))) float    v2f;
typedef __attribute__((ext_vector_type(8))) float    v8f;
typedef __attribute__((ext_vector_type(4))) unsigned u32x4;
typedef __attribute__((ext_vector_type(8))) unsigned u32x8;

__device__ __forceinline__ v8f wmma4(v2f a, v2f b, v8f c) {
  // D = A(16x4) * B(4x16) + C, full fp32.
  return __builtin_amdgcn_wmma_f32_16x16x4_f32(
      /*neg_a=*/false, a, /*neg_b=*/false, b,
      /*c_mod=*/(short)0, c, /*reuse_a=*/false, /*reuse_b=*/false);
}

// Issue one TDM descriptor: contiguous 1-D copy of `ndwords` DWORDs from
// global to LDS.  pad128: insert 1 DWORD of LDS padding every 128 DWORDs
// (turns 128-DWORD rows into stride-129 rows -> bank-conflict-free columns).
// D# layout per CDNA5 ISA ch.8 (group0 128b + group1 256b; groups 2/3 NULL).
__device__ __forceinline__ void tdm_load_1d(unsigned lds_addr, const void* gptr,
                                            unsigned ndwords, bool pad128) {
  unsigned long long ga = (unsigned long long)gptr;
  u32x4 g0;
  g0[0] = 1u;                                          // count=1 (valid user D#)
  g0[1] = lds_addr;                                    // lds_addr [63:32]
  g0[2] = (unsigned)(ga & 0xffffffffull);              // global_addr lo
  g0[3] = (unsigned)((ga >> 32) & 0x01ffffffull)       // global_addr [56:32]
          | (2u << 30);                                // type=2 ("image")
  u32x8 g1;
  g1[0] = (2u << 16)                                   // data_size=2 -> 4 bytes
          | (pad128 ? ((1u << 20) | (6u << 22)) : 0u); // pad_enable, interval=128dw, amount=1dw
  g1[1] = (ndwords & 0xffffu) << 16;                   // tensor_dim0[15:0]
  g1[2] = ((ndwords >> 16) & 0xffffu) | (1u << 16);    // tensor_dim0[31:16] | tensor_dim1=1
  g1[3] = (ndwords & 0xffffu) << 16;                   // tensor_dim1 hi=0 | tile_dim0
  g1[4] = 0u;                                          // tile_dim1=0, tile_dim2=0 (unused)
  g1[5] = ndwords;                                     // tensor_dim0_stride lo
  g1[6] = 0u;                                          // stride0 hi | stride1 lo
  g1[7] = 0u;                                          // stride1 hi
  asm volatile("tensor_load_to_lds %0, %1" :: "s"(g0), "s"(g1) : "memory");
}

// ---- kernel 1: v = x + 0.5*(roll1(x)+roll2(x));  per-node LN half-sums -----
__global__ void prep_v_stats(const float* __restrict__ x, float* __restrict__ v,
                             float* __restrict__ su, float* __restrict__ squ,
                             float* __restrict__ sv, float* __restrict__ sqv) {
  int p = blockIdx.x;                 // p = b*64 + t
  int b = p >> 6, t = p & 63;
  int t1 = (t + 1) & 63, t2 = (t + 2) & 63;
  const float* x0  = x + (size_t)p * 4096;
  const float* xr1 = x + (size_t)(b * 64 + t1) * 4096;
  const float* xr2 = x + (size_t)(b * 64 + t2) * 4096;
  __shared__ float vs[4096];
  for (int idx = threadIdx.x; idx < 4096; idx += 256) {
    float val = x0[idx] + 0.5f * (xr1[idx] + xr2[idx]);
    vs[idx] = val;
    v[(size_t)p * 4096 + idx] = val;
  }
  __syncthreads();
  if (threadIdx.x < 64) {
    int n = threadIdx.x;
    float s_u = 0.f, q_u = 0.f, s_v = 0.f, q_v = 0.f;
    for (int d = 0; d < 64; ++d) {
      float xu = 2.0f * x0[n * 64 + d];
      s_u += xu; q_u += xu * xu;
      float xv = vs[n * 64 + d];
      s_v += xv; q_v += xv * xv;
    }
    su[p * 64 + n] = s_u;  squ[p * 64 + n] = q_u;
    sv[p * 64 + n] = s_v;  sqv[p * 64 + n] = q_v;
  }
}

// ---- kernel 2: fold ln2_g into W (transposed), row-sums c, fused bias ------
__global__ void prep_weights(const float* __restrict__ W_w,
                             const float* __restrict__ g2,
                             const float* __restrict__ b2,
                             const float* __restrict__ W_b,
                             float* __restrict__ WpT, float* __restrict__ c,
                             float* __restrict__ bp) {
  int idx = blockIdx.x * 256 + threadIdx.x;      // 0..16383
  int f = idx >> 7, g = idx & 127;
  WpT[idx] = W_w[g * 128 + f] * g2[f];           // WpT[f][g] = W'[g][f]
  if (blockIdx.x == 0 && threadIdx.x < 128) {
    int gg = threadIdx.x;
    float cs = 0.f, bs = W_b[gg];
    for (int ff = 0; ff < 128; ++ff) {
      float w = W_w[gg * 128 + ff];
      cs += w * g2[ff];
      bs += w * b2[ff];
    }
    c[gg] = cs;  bp[gg] = bs;
  }
}

// ---- kernel 3: AU = (2x)*A'^T, BV = v*B'^T  via V_WMMA_F32_16X16X4_F32 -----
__global__ void gemm_uv(const float* __restrict__ x, const float* __restrict__ v,
                        const float* __restrict__ WpT,
                        float* __restrict__ AU, float* __restrict__ BV) {
  int p = blockIdx.x;
  int lane = threadIdx.x & 31, wave = threadIdx.x >> 5;
  int hi = lane >> 4, lo = lane & 15;
  for (int tt = wave; tt < 64; tt += 8) {        // 32 AU tiles + 32 BV tiles
    int isBV = tt >> 5;
    int t = tt & 31;
    int m0 = (t & 3) * 16;                       // node-row tile  (M=64 -> 4)
    int g0 = (t >> 2) * 16;                      // feature tile   (N=128 -> 8)
    const float* src = (isBV ? v : x) + (size_t)p * 4096;
    float ascale = isBV ? 1.0f : 2.0f;
    int foff = isBV ? 64 : 0;
    int row = m0 + lo;
    int g = g0 + lo;
    v8f acc = {};
    for (int kk = 0; kk < 64; kk += 4) {
      int k = kk + hi * 2;
      v2f a, bf;
      a.x = ascale * src[row * 64 + k];
      a.y = ascale * src[row * 64 + k + 1];
      bf.x = WpT[(foff + k) * 128 + g];
      bf.y = WpT[(foff + k + 1) * 128 + g];
      acc = wmma4(a, bf, acc);
    }
    float* dst = (isBV ? BV : AU) + (size_t)p * 8192;
#pragma unroll
    for (int r = 0; r < 8; ++r)
      dst[(m0 + r + hi * 8) * 128 + g] = acc[r];
  }
}

// ---- kernel 4: per-pair LN-combine + leaky_relu + dot(a_w) -> adj ----------
__global__ void pair_scores(const float* __restrict__ AU, const float* __restrict__ BV,
                            const float* __restrict__ su, const float* __restrict__ squ,
                            const float* __restrict__ sv, const float* __restrict__ sqv,
                            const float* __restrict__ c,  const float* __restrict__ bp,
                            const float* __restrict__ a_w, const float* __restrict__ a_b,
                            float* __restrict__ adj) {
  int p  = blockIdx.x >> 1;
  int ih = blockIdx.x & 1;                       // which 32-row i half
  __shared__ float AU_s[32 * 128];
  __shared__ float BV_s[64 * 129];               // stride-129 rows via TDM padding
  __shared__ float su_s[32], squ_s[32], sv_s[64], sqv_s[64];
  __shared__ float c_s[128], bp_s[128], aw_s[128];
  const float* AUp = AU + (size_t)p * 8192 + (size_t)ih * 4096;
  const float* BVp = BV + (size_t)p * 8192;
  if (threadIdx.x < 32) {                        // wave 0: TDM global->LDS copies
    tdm_load_1d((unsigned)(size_t)AU_s, AUp, 4096, false);
    tdm_load_1d((unsigned)(size_t)BV_s, BVp, 8192, true);   // +1dw pad / 128dw
  }
  if (threadIdx.x < 32) {
    su_s[threadIdx.x]  = su[p * 64 + ih * 32 + threadIdx.x];
    squ_s[threadIdx.x] = squ[p * 64 + ih * 32 + threadIdx.x];
  } else if (threadIdx.x < 96) {
    int j = threadIdx.x - 32;
    sv_s[j] = sv[p * 64 + j];  sqv_s[j] = sqv[p * 64 + j];
  }
  if (threadIdx.x < 128) {
    c_s[threadIdx.x]  = c[threadIdx.x];
    bp_s[threadIdx.x] = bp[threadIdx.x];
    aw_s[threadIdx.x] = a_w[threadIdx.x];
  }
  if (threadIdx.x < 32)                          // wave 0 waits for TDM...
    __builtin_amdgcn_s_wait_tensorcnt(0);
  __syncthreads();                               // ...then barrier publishes LDS
  float ab = a_b[0];
  int j = threadIdx.x & 63;                      // lane-consecutive j
  int grp = threadIdx.x >> 6;                    // wave-uniform i group
  for (int ii = 0; ii < 8; ++ii) {
    int il = grp * 8 + ii;                       // local i in [0,32)
    int i  = ih * 32 + il;
    float mu  = (su_s[il]  + sv_s[j])  * (1.0f / 128.0f);
    float ex2 = (squ_s[il] + sqv_s[j]) * (1.0f / 128.0f);
    float inv = rsqrtf(ex2 - mu * mu + 1e-5f);
    float acc = ab;
    const float* au = AU_s + il * 128;           // uniform addr -> LDS broadcast
    const float* bv = BV_s + j * 129;            // conflict-free
#pragma unroll 8
    for (int k = 0; k < 128; ++k) {
      float tk = fmaf(-mu, c_s[k], au[k] + bv[k]) * inv + bp_s[k];
      float lr = tk > 0.0f ? tk : 0.01f * tk;
      acc = fmaf(lr, aw_s[k], acc);
    }
    adj[(size_t)p * 4096 + i * 64 + j] = acc;
  }
}

// ---- kernel 5: softmax, h'=att@x (WMMA), LN1, conv1(WMMA)+relu, conv2 ------
__global__ void attn_out(const float* __restrict__ x, const float* __restrict__ adj,
                         const float* __restrict__ ln1_g, const float* __restrict__ ln1_b,
                         const float* __restrict__ c1w, const float* __restrict__ c1b,
                         const float* __restrict__ c2w, const float* __restrict__ c2b,
                         float* __restrict__ out) {
  int p = blockIdx.x;
  __shared__ float smem[4160 + 4096 + 4096];
  float* att = smem;                 // 64x65; later reused as ys (LN output)
  float* xs  = smem + 4160;          // 64x64; later reused as zs (relu output)
  float* hp  = smem + 4160 + 4096;   // 64x64
  const float* xp = x + (size_t)p * 4096;
  if (threadIdx.x < 32)              // wave 0: TDM copy of x tile (overlaps softmax)
    tdm_load_1d((unsigned)(size_t)xs, xp, 4096, false);
  if (threadIdx.x < 64) {            // row-wise softmax over j (reads adj + att only)
    int i = threadIdx.x;
    const float* er = adj + (size_t)p * 4096 + i * 64;
    float m = -3.4e38f;
    for (int k = 0; k < 64; ++k) m = fmaxf(m, er[k]);
    float s = 0.f;
    for (int k = 0; k < 64; ++k) { float e = __expf(er[k] - m); att[i * 65 + k] = e; s += e; }
    float is = 1.0f / s;
    for (int k = 0; k < 64; ++k) att[i * 65 + k] *= is;
  }
  if (threadIdx.x < 32)
    __builtin_amdgcn_s_wait_tensorcnt(0);
  __syncthreads();
  int lane = threadIdx.x & 31, wave = threadIdx.x >> 5;
  int hi = lane >> 4, lo = lane & 15;
  // GEMM1: hp = att(64x64) @ xs(64x64), 16 tiles, 2 per wave
  for (int q = 0; q < 2; ++q) {
    int tt = wave * 2 + q;
    int m0 = (tt >> 2) * 16, n0 = (tt & 3) * 16;
    v8f acc = {};
    for (int kk = 0; kk < 64; kk += 4) {
      int k = kk + hi * 2;
      v2f a, bf;
      a.x = att[(m0 + lo) * 65 + k];
      a.y = att[(m0 + lo) * 65 + k + 1];
      bf.x = xs[k * 64 + n0 + lo];
      bf.y = xs[(k + 1) * 64 + n0 + lo];
      acc = wmma4(a, bf, acc);
    }
#pragma unroll
    for (int r = 0; r < 8; ++r)
      hp[(m0 + r + hi * 8) * 64 + n0 + lo] = acc[r];
  }
  __syncthreads();
  // LN1 over (hp + xs) rows -> ys (reuse att buffer, stride 65)
  if (threadIdx.x < 64) {
    int n = threadIdx.x;
    float s = 0.f;
    for (int d = 0; d < 64; ++d) s += hp[n * 64 + d] + xs[n * 64 + d];
    float mu = s * (1.0f / 64.0f);
    float q = 0.f;
    for (int d = 0; d < 64; ++d) {
      float dv = hp[n * 64 + d] + xs[n * 64 + d] - mu;
      q += dv * dv;
    }
    float inv = rsqrtf(q * (1.0f / 64.0f) + 1e-5f);
    for (int d = 0; d < 64; ++d) {
      float dv = (hp[n * 64 + d] + xs[n * 64 + d] - mu) * inv;
      att[n * 65 + d] = dv * ln1_g[d] + ln1_b[d];    // ys
    }
  }
  __syncthreads();
  // GEMM2: zs = relu(ys @ conv1_w^T + c1b), zs reuses xs buffer (stride 64)
  for (int q = 0; q < 2; ++q) {
    int tt = wave * 2 + q;
    int m0 = (tt >> 2) * 16, n0 = (tt & 3) * 16;
    v8f acc = {};
    for (int kk = 0; kk < 64; kk += 4) {
      int k = kk + hi * 2;
      v2f a, bf;
      a.x = att[(m0 + lo) * 65 + k];                 // ys
      a.y = att[(m0 + lo) * 65 + k + 1];
      bf.x = c1w[(n0 + lo) * 64 + k];                // B[d][e] = conv1_w[e][d]
      bf.y = c1w[(n0 + lo) * 64 + k + 1];
      acc = wmma4(a, bf, acc);
    }
    int g = n0 + lo;
    float cb = c1b[g];
#pragma unroll
    for (int r = 0; r < 8; ++r)
      xs[(m0 + r + hi * 8) * 64 + g] = fmaxf(acc[r] + cb, 0.0f);  // zs
  }
  __syncthreads();
  if (threadIdx.x < 64) {            // conv2 (out_channels=1) + transpose store
    int n = threadIdx.x;
    float y2 = c2b[0];
    for (int g = 0; g < 64; ++g) y2 = fmaf(xs[n * 64 + g], c2w[g], y2);
    int b = p >> 6, t = p & 63;
    out[(size_t)b * 4096 + n * 64 + t] = y2;         // out[b][n][t]
  }
}

// ---------------------------------------------------------------------------
extern "C" void kernel_launch(void* const* d_in, const int* in_sizes, int n_in,
                              void* d_out, int out_size, void* d_ws, size_t ws_size,
                              hipStream_t stream) {
  (void)in_sizes; (void)n_in; (void)out_size; (void)ws_size;
  const float* x     = (const float*)d_in[0];
  const float* ln2_g = (const float*)d_in[1];
  const float* ln2_b = (const float*)d_in[2];
  const float* W_w   = (const float*)d_in[3];
  const float* W_b   = (const float*)d_in[4];
  const float* a_w   = (const float*)d_in[5];
  const float* a_b   = (const float*)d_in[6];
  const float* ln1_g = (const float*)d_in[7];
  const float* ln1_b = (const float*)d_in[8];
  const float* c1w   = (const float*)d_in[9];
  const float* c1b   = (const float*)d_in[10];
  const float* c2w   = (const float*)d_in[11];
  const float* c2b   = (const float*)d_in[12];

  float* out = (float*)d_out;
  float* adj = out + 8192;

  float* ws  = (float*)d_ws;
  float* v   = ws;                 // 524288
  float* su  = v   + 524288;       // 8192
  float* squ = su  + 8192;         // 8192
  float* sv  = squ + 8192;         // 8192
  float* sqv = sv  + 8192;         // 8192
  float* WpT = sqv + 8192;         // 16384
  float* c   = WpT + 16384;        // 128
  float* bp  = c   + 128;          // 128
  float* AU  = bp  + 128;          // 1048576
  float* BV  = AU  + 1048576;      // 1048576

  prep_v_stats<<<128, 256, 0, stream>>>(x, v, su, squ, sv, sqv);
  prep_weights<<<64, 256, 0, stream>>>(W_w, ln2_g, ln2_b, W_b, WpT, c, bp);
  gemm_uv<<<128, 256, 0, stream>>>(x, v, WpT, AU, BV);
  pair_scores<<<256, 256, 0, stream>>>(AU, BV, su, squ, sv, sqv, c, bp, a_w, a_b, adj);
  attn_out<<<128, 256, 0, stream>>>(x, adj, ln1_g, ln1_b, c1w, c1b, c2w, c2b, out);
}